// ComplexPtreeLayer_42116449304576
// MI455X (gfx1250) — compile-verified
//
#include <hip/hip_runtime.h>
#include <math.h>

// ---------------------------------------------------------------------------
// ComplexPtreeLayer for MI455X (gfx1250): fp32 WMMA (V_WMMA_F32_16X16X4_F32)
// keeps exact fp32 reference semantics; gathers staged in LDS with padded
// stride (260 words) for bank-conflict-free A-fragment ds_read_b64.
// ---------------------------------------------------------------------------

typedef float v2f __attribute__((ext_vector_type(2)));
typedef float v8f __attribute__((ext_vector_type(8)));

#define HDIM 256
#define KCH 4
#define GSTRIDE 260  // padded LDS row stride in words (260 % 4 == 0 -> 16B stores OK)

// ---------------------------------------------------------------------------
// acc += A(16 x 256, LDS row-major stride GSTRIDE) * W[n0:n0+16, :]^T
// A frag (16x4 f32): lanes 0-15 hold M=lane, K={0,1}; lanes 16-31 M=lane-16,
// K={2,3}. B frag (4x16): lanes 0-15 K={0,1}, lanes 16-31 K={2,3}, N=lane%16.
// C/D frag: VGPR v -> M = v + 8*(lane>=16), N = lane%16.
// ---------------------------------------------------------------------------
__device__ __forceinline__ v8f gemm_tile_ldsA(const float* __restrict__ Als,
                                              const float* __restrict__ W,
                                              int n0, int lane, v8f acc) {
  const int mrow = lane & 15;
  const int kb = (lane >> 4) << 1;  // 0 or 2
  const int nl = lane & 15;
  const float* __restrict__ arow = Als + mrow * GSTRIDE + kb;
  const float* __restrict__ brow = W + (size_t)(n0 + nl) * HDIM + kb;
#pragma unroll 4
  for (int kk = 0; kk < HDIM; kk += 4) {
    v2f a = *(const v2f*)(arow + kk);
    v2f b = *(const v2f*)(brow + kk);
    acc = __builtin_amdgcn_wmma_f32_16x16x4_f32(false, a, false, b, (short)0,
                                                acc, false, false);
  }
  return acc;
}

// ---------------------------------------------------------------------------
// data0 = x[initial_map]
// ---------------------------------------------------------------------------
__global__ void gather_kernel(const float* __restrict__ x,
                              const int* __restrict__ map,
                              float* __restrict__ out, int n) {
  int t = blockIdx.x * blockDim.x + threadIdx.x;  // one float4 per thread
  int row = t >> 6;                               // H/4 = 64 chunks per row
  int c4 = t & 63;
  if (row < n) {
    const float4* src = (const float4*)(x + (size_t)map[row] * HDIM);
    float4* dst = (float4*)(out + (size_t)row * HDIM);
    dst[c4] = src[c4];
  }
}

// ---------------------------------------------------------------------------
// One layer: temp[m,H] from data[prev,H] via gather + typed GEMMs + select.
// WG = 16 nodes, 256 threads = 8 wave32, each wave owns N-tiles {w, w+8}.
// ---------------------------------------------------------------------------
__launch_bounds__(256)
__global__ void layer_kernel(const float* __restrict__ data,  // [prev,H]
                             const int* __restrict__ om,      // [K,m]
                             const int* __restrict__ tmask,   // [m]
                             const float* __restrict__ Wz,    // [K,H,H]
                             const float* __restrict__ Wzi,   // [H,H]
                             const float* __restrict__ Ws,    // [K,H,H]
                             const float* __restrict__ Wsi,   // [H,H]
                             const float* __restrict__ Wp,    // [H,H]
                             float* __restrict__ temp,        // [m,H]
                             int m) {
  __shared__ float G[KCH * 16 * GSTRIDE];  // gathered children
  __shared__ float U[16 * GSTRIDE];        // intermediate (z then s)
  __shared__ int tms[16];
  __shared__ int aes[16];

  const int tid = threadIdx.x;
  const int wave = tid >> 5;
  const int lane = tid & 31;
  const int i0 = blockIdx.x * 16;

  if (tid < 16) {
    tms[tid] = tmask[i0 + tid];
    int ae = 1;
#pragma unroll
    for (int k = 0; k < KCH; ++k) ae &= (om[(size_t)k * m + i0 + tid] < 0);
    aes[tid] = ae;
  }

  // Stage gathered children rows into LDS (zero row for missing child).
  for (int e = tid; e < KCH * 16 * 64; e += 256) {
    int k = e >> 10;          // 16*64 chunks per child
    int i = (e >> 6) & 15;
    int c4 = e & 63;
    int child = om[(size_t)k * m + i0 + i];
    float4 val = make_float4(0.f, 0.f, 0.f, 0.f);
    if (child >= 0)
      val = ((const float4*)(data + (size_t)child * HDIM))[c4];
    *(float4*)&G[(k * 16 + i) * GSTRIDE + c4 * 4] = val;
  }
  __syncthreads();

  const v8f vzero = {0.f, 0.f, 0.f, 0.f, 0.f, 0.f, 0.f, 0.f};
  const int col16 = lane & 15;
  const int rbase = (lane >> 4) * 8;

  v8f zacc[2], sacc[2], pacc[2];

  // ---- Z path: U = sum_k G_k @ Wz[k]^T ; Z = U @ Wzi^T ----
  {
    v8f uacc[2];
#pragma unroll
    for (int t = 0; t < 2; ++t) {
      int n0 = (wave + 8 * t) * 16;
      v8f acc = vzero;
#pragma unroll
      for (int k = 0; k < KCH; ++k)
        acc = gemm_tile_ldsA(&G[k * 16 * GSTRIDE], Wz + (size_t)k * HDIM * HDIM,
                             n0, lane, acc);
      uacc[t] = acc;
    }
#pragma unroll
    for (int t = 0; t < 2; ++t) {
      int n0 = (wave + 8 * t) * 16;
#pragma unroll
      for (int v = 0; v < 8; ++v)
        U[(rbase + v) * GSTRIDE + n0 + col16] = uacc[t][v];
    }
    __syncthreads();
#pragma unroll
    for (int t = 0; t < 2; ++t)
      zacc[t] = gemm_tile_ldsA(U, Wzi, (wave + 8 * t) * 16, lane, vzero);
    __syncthreads();  // all reads of U done before s-path overwrites it
  }

  // ---- S path: U = sum_k G_k @ Ws[k]^T ; S = U @ Wsi^T ----
  {
    v8f uacc[2];
#pragma unroll
    for (int t = 0; t < 2; ++t) {
      int n0 = (wave + 8 * t) * 16;
      v8f acc = vzero;
#pragma unroll
      for (int k = 0; k < KCH; ++k)
        acc = gemm_tile_ldsA(&G[k * 16 * GSTRIDE], Ws + (size_t)k * HDIM * HDIM,
                             n0, lane, acc);
      uacc[t] = acc;
    }
#pragma unroll
    for (int t = 0; t < 2; ++t) {
      int n0 = (wave + 8 * t) * 16;
#pragma unroll
      for (int v = 0; v < 8; ++v)
        U[(rbase + v) * GSTRIDE + n0 + col16] = uacc[t][v];
    }
    __syncthreads();
#pragma unroll
    for (int t = 0; t < 2; ++t)
      sacc[t] = gemm_tile_ldsA(U, Wsi, (wave + 8 * t) * 16, lane, vzero);
  }

  // ---- P path: P = G_0 @ Wp^T (G untouched since load) ----
#pragma unroll
  for (int t = 0; t < 2; ++t)
    pacc[t] = gemm_tile_ldsA(&G[0], Wp, (wave + 8 * t) * 16, lane, vzero);

  // ---- per-row type selection + store ----
#pragma unroll
  for (int t = 0; t < 2; ++t) {
    int n0 = (wave + 8 * t) * 16;
    int col = n0 + col16;
#pragma unroll
    for (int v = 0; v < 8; ++v) {
      int i = rbase + v;
      int tm = tms[i];
      float val = G[i * GSTRIDE + col];  // default: gathered[0] (type 0)
      if (tm == 1) val = pacc[t][v];
      else if (tm == 2) val = zacc[t][v];
      else if (tm == 3) val = sacc[t][v];
      if (aes[i]) val = 0.f;
      temp[(size_t)(i0 + i) * HDIM + col] = val;
    }
  }
}

// ---------------------------------------------------------------------------
// Contiguous segment-sum + segment-max(tmask) + ELU. One block per segment.
// ---------------------------------------------------------------------------
__global__ void pool_kernel(const float* __restrict__ temp,
                            const int* __restrict__ tmask,
                            float* __restrict__ pooled, int* __restrict__ tm2,
                            int S) {
  int seg = blockIdx.x;
  int col = threadIdx.x;  // 256 cols
  float s = 0.f;
  int tm = 0;
  for (int j = 0; j < S; ++j) {
    s += temp[(size_t)(seg * S + j) * HDIM + col];
    int t = tmask[seg * S + j];
    tm = t > tm ? t : tm;
  }
  if (tm != 0) s = s > 0.f ? s : expm1f(s);  // ELU, alpha=1
  pooled[(size_t)seg * HDIM + col] = s;
  if (col == 0) tm2[seg] = tm;
}

// ---------------------------------------------------------------------------
// Conditional pooled @ Wpf^T for rows with tm2==1. Safe in-place (rows staged
// in LDS before any write).
// ---------------------------------------------------------------------------
__launch_bounds__(256)
__global__ void pf_kernel(const float* __restrict__ src,
                          const int* __restrict__ tm2,
                          const float* __restrict__ Wpf,
                          float* __restrict__ dst) {
  __shared__ float X[16 * GSTRIDE];
  __shared__ int flg[16];
  const int tid = threadIdx.x;
  const int wave = tid >> 5;
  const int lane = tid & 31;
  const int i0 = blockIdx.x * 16;

  if (tid < 16) flg[tid] = tm2[i0 + tid];
  for (int e = tid; e < 16 * 64; e += 256) {
    int i = e >> 6, c4 = e & 63;
    *(float4*)&X[i * GSTRIDE + c4 * 4] =
        ((const float4*)(src + (size_t)(i0 + i) * HDIM))[c4];
  }
  __syncthreads();

  const v8f vzero = {0.f, 0.f, 0.f, 0.f, 0.f, 0.f, 0.f, 0.f};
  const int col16 = lane & 15;
  const int rbase = (lane >> 4) * 8;
#pragma unroll
  for (int t = 0; t < 2; ++t) {
    int n0 = (wave + 8 * t) * 16;
    v8f acc = gemm_tile_ldsA(X, Wpf, n0, lane, vzero);
    int col = n0 + col16;
#pragma unroll
    for (int v = 0; v < 8; ++v) {
      int i = rbase + v;
      float val = (flg[i] == 1) ? acc[v] : X[i * GSTRIDE + col];
      dst[(size_t)(i0 + i) * HDIM + col] = val;
    }
  }
}

// ---------------------------------------------------------------------------
extern "C" void kernel_launch(void* const* d_in, const int* in_sizes, int n_in,
                              void* d_out, int out_size, void* d_ws,
                              size_t ws_size, hipStream_t stream) {
  const float* x = (const float*)d_in[0];
  const int* imap = (const int*)d_in[1];
  const int* om[3] = {(const int*)d_in[2], (const int*)d_in[5],
                      (const int*)d_in[8]};
  const int* tmk[3] = {(const int*)d_in[4], (const int*)d_in[7],
                       (const int*)d_in[10]};
  const float* Wz = (const float*)d_in[11];
  const float* Wzi = (const float*)d_in[12];
  const float* Ws = (const float*)d_in[13];
  const float* Wsi = (const float*)d_in[14];
  const float* Wp = (const float*)d_in[15];
  const float* Wpf = (const float*)d_in[16];

  float* bufA = (float*)d_ws;                    // 64 MB: layer input / pooled
  float* bufB = bufA + (size_t)65536 * HDIM;     // 64 MB: temp
  int* tm2 = (int*)(bufB + (size_t)65536 * HDIM);

  const int ms[3] = {65536, 32768, 8192};
  const int gs[3] = {32768, 8192, 2048};

  // data0 = x[initial_map]
  gather_kernel<<<(65536 * 64) / 256, 256, 0, stream>>>(x, imap, bufA, 65536);

  for (int l = 0; l < 3; ++l) {
    int m = ms[l], g = gs[l], S = m / g;
    layer_kernel<<<m / 16, 256, 0, stream>>>(bufA, om[l], tmk[l], Wz, Wzi, Ws,
                                             Wsi, Wp, bufB, m);
    pool_kernel<<<g, 256, 0, stream>>>(bufB, tmk[l], bufA, tm2, S);
    float* dst = (l == 2) ? (float*)d_out : bufA;
    pf_kernel<<<g / 16, 256, 0, stream>>>(bufA, tm2, Wpf, dst);
  }
}